// FBNetGenEnhanced_81784767250571
// MI455X (gfx1250) — compile-verified
//
#include <hip/hip_runtime.h>
#include <math.h>

#define BQ 8
#define NN 268
#define INDIM 268
#define HID 128
#define HEADS 4
#define OUTC 32
#define GHD 64
#define ROWS (BQ * NN)          // 2144 (exactly 134 tiles of 16)
#define SQRT_NN 16.3707055437f  // sqrt(268)
#define NEG_SLOPE 0.2f

typedef float v2f __attribute__((ext_vector_type(2)));
typedef float v8f __attribute__((ext_vector_type(8)));

__device__ __forceinline__ v8f wmma_f32_16x16x4(v2f a, v2f b, v8f c) {
  // D = A(16x4,f32) * B(4x16,f32) + C(16x16,f32), wave32
  return __builtin_amdgcn_wmma_f32_16x16x4_f32(false, a, false, b, (short)0, c,
                                               false, false);
}

// ---------------------------------------------------------------------------
// Generic strided GEMM via V_WMMA_F32_16X16X4_F32.
//   C(m,n) = act( sum_k A(m,k)*B(k,n) + bias[n] )
//   A(m,k) = A[m*sAm + k*sAk];  B(k,n) = Bp[k*sBk + n*sBn]
// One wave (32 threads) per 16x16 output tile. K % 4 == 0, Nn % 16 == 0.
// act: 0 = none, 1 = relu, 2 = tanh
// ---------------------------------------------------------------------------
__global__ void gemm_wmma_kernel(const float* __restrict__ A, int sAm, int sAk,
                                 const float* __restrict__ Bp, int sBk, int sBn,
                                 const float* __restrict__ bias,
                                 float* __restrict__ C, int sCm,
                                 int M, int K, int act) {
  const int lane = threadIdx.x & 31;
  const int lm = lane & 15;
  const int hi = lane >> 4;
  const int baseM = blockIdx.y << 4;
  const int baseN = blockIdx.x << 4;
  const int m = baseM + lm;
  const int mc = (m < M) ? m : (M - 1);
  const bool mOK = (m < M);
  const int n = baseN + lm;

  v8f acc = {};
  for (int k0 = 0; k0 < K; k0 += 4) {
    const int ka = k0 + (hi << 1);
    float a0 = A[(long)mc * sAm + (long)ka * sAk];
    float a1 = A[(long)mc * sAm + (long)(ka + 1) * sAk];
    a0 = mOK ? a0 : 0.0f;
    a1 = mOK ? a1 : 0.0f;
    v2f av;
    av.x = a0;
    av.y = a1;
    v2f bv;
    bv.x = Bp[(long)ka * sBk + (long)n * sBn];
    bv.y = Bp[(long)(ka + 1) * sBk + (long)n * sBn];
    acc = wmma_f32_16x16x4(av, bv, acc);
  }
#pragma unroll
  for (int r = 0; r < 8; ++r) {
    const int mr = baseM + r + (hi << 3);
    if (mr < M) {
      float v = acc[r];
      if (bias) v += bias[n];
      if (act == 1) v = fmaxf(v, 0.0f);
      else if (act == 2) v = tanhf(v);
      C[(long)mr * sCm + n] = v;
    }
  }
}

// ---------------------------------------------------------------------------
// Attention aggregation: out[g,i,h*32+c] = sum_j alpha[g,i,j,h]*xl[g,j,h*32+c]
//                                          + bias[h*32+c]
// grid: x = 2 (c tiles), y = 17 (i tiles of 16), z = g*4+h. One wave/block.
// ---------------------------------------------------------------------------
__global__ void attn_agg_wmma_kernel(const float* __restrict__ alpha,
                                     const float* __restrict__ xl,
                                     const float* __restrict__ bias,
                                     float* __restrict__ outb) {
  const int lane = threadIdx.x & 31;
  const int lm = lane & 15;
  const int hi = lane >> 4;
  const int gh = blockIdx.z;
  const int g = gh >> 2;
  const int h = gh & 3;
  const int baseM = blockIdx.y << 4;
  const int baseN = blockIdx.x << 4;
  const int m = baseM + lm;
  const int mc = (m < NN) ? m : (NN - 1);
  const bool mOK = (m < NN);
  const int n = baseN + lm;

  const float* Ag = alpha + (long)g * NN * NN * HEADS + h;
  const float* Bg = xl + (long)g * NN * HID + h * OUTC;

  v8f acc = {};
  for (int k0 = 0; k0 < NN; k0 += 4) {  // 268 % 4 == 0
    const int ka = k0 + (hi << 1);
    float a0 = Ag[((long)mc * NN + ka) * HEADS];
    float a1 = Ag[((long)mc * NN + ka + 1) * HEADS];
    a0 = mOK ? a0 : 0.0f;
    a1 = mOK ? a1 : 0.0f;
    v2f av;
    av.x = a0;
    av.y = a1;
    v2f bv;
    bv.x = Bg[(long)ka * HID + n];
    bv.y = Bg[(long)(ka + 1) * HID + n];
    acc = wmma_f32_16x16x4(av, bv, acc);
  }
#pragma unroll
  for (int r = 0; r < 8; ++r) {
    const int mr = baseM + r + (hi << 3);
    if (mr < NN) {
      outb[((long)g * NN + mr) * HID + h * OUTC + n] =
          acc[r] + bias[h * OUTC + n];
    }
  }
}

// ---------------------------------------------------------------------------
// LayerNorm (+optional relu) over rows of 128. One block (128 thr) per row.
// ---------------------------------------------------------------------------
__global__ void lnrelu_kernel(const float* __restrict__ X,
                              const float* __restrict__ g,
                              const float* __restrict__ b,
                              float* __restrict__ Y, int relu) {
  const int row = blockIdx.x;
  const int tid = threadIdx.x;
  __shared__ float red[HID];
  float v = X[(long)row * HID + tid];
  red[tid] = v;
  __syncthreads();
  for (int s = HID / 2; s > 0; s >>= 1) {
    if (tid < s) red[tid] += red[tid + s];
    __syncthreads();
  }
  const float mean = red[0] * (1.0f / HID);
  __syncthreads();
  const float d = v - mean;
  red[tid] = d * d;
  __syncthreads();
  for (int s = HID / 2; s > 0; s >>= 1) {
    if (tid < s) red[tid] += red[tid + s];
    __syncthreads();
  }
  const float var = red[0] * (1.0f / HID);
  float o = d * rsqrtf(var + 1e-5f) * g[tid] + b[tid];
  if (relu) o = fmaxf(o, 0.0f);
  Y[(long)row * HID + tid] = o;
}

// ---------------------------------------------------------------------------
// Column mean over nodes for PairNorm. grid = BQ, block = 128.
// ---------------------------------------------------------------------------
__global__ void colmean_kernel(const float* __restrict__ outb,
                               float* __restrict__ colmean) {
  const int g = blockIdx.x;
  const int d = threadIdx.x;
  float s = 0.0f;
  for (int i = 0; i < NN; ++i) s += outb[((long)g * NN + i) * HID + d];
  colmean[g * HID + d] = s * (1.0f / NN);
}

// ---------------------------------------------------------------------------
// Fused PairNorm + residual + LayerNorm (+optional relu). Block/row, 128 thr.
// ---------------------------------------------------------------------------
__global__ void pn_res_ln_kernel(const float* __restrict__ outb,
                                 const float* __restrict__ colmean,
                                 const float* __restrict__ ident,
                                 const float* __restrict__ lng,
                                 const float* __restrict__ lnb,
                                 float* __restrict__ xh, int relu) {
  const int row = blockIdx.x;
  const int g = row / NN;
  const int tid = threadIdx.x;
  __shared__ float red[HID];
  float v = outb[(long)row * HID + tid] - colmean[g * HID + tid];
  red[tid] = v * v;
  __syncthreads();
  for (int s = HID / 2; s > 0; s >>= 1) {
    if (tid < s) red[tid] += red[tid + s];
    __syncthreads();
  }
  const float nrm = sqrtf(red[0]) + 1e-6f;
  __syncthreads();
  v = v / nrm * SQRT_NN;
  v += ident[(long)row * HID + tid];
  // LayerNorm
  red[tid] = v;
  __syncthreads();
  for (int s = HID / 2; s > 0; s >>= 1) {
    if (tid < s) red[tid] += red[tid + s];
    __syncthreads();
  }
  const float mean = red[0] * (1.0f / HID);
  __syncthreads();
  const float d = v - mean;
  red[tid] = d * d;
  __syncthreads();
  for (int s = HID / 2; s > 0; s >>= 1) {
    if (tid < s) red[tid] += red[tid + s];
    __syncthreads();
  }
  const float var = red[0] * (1.0f / HID);
  float o = d * rsqrtf(var + 1e-5f) * lng[tid] + lnb[tid];
  if (relu) o = fmaxf(o, 0.0f);
  xh[(long)row * HID + tid] = o;
}

// ---------------------------------------------------------------------------
// GATv2 logits: logits[g,i,j,h] = sum_c att[h,c]*lrelu(xr[g,i,hc]+xl[g,j,hc])
// grid = ROWS (g*NN+i), block = 256.
// ---------------------------------------------------------------------------
__global__ void logits_kernel(const float* __restrict__ xl,
                              const float* __restrict__ xr,
                              const float* __restrict__ att,
                              float* __restrict__ logits) {
  const int row = blockIdx.x;  // g*NN + i
  const int g = row / NN;
  const int tid = threadIdx.x;
  __shared__ float sxr[HID];
  __shared__ float satt[HID];
  if (tid < HID) {
    sxr[tid] = xr[(long)row * HID + tid];
    satt[tid] = att[tid];
  }
  __syncthreads();
  for (int j = tid; j < NN; j += blockDim.x) {
    const float* xlr = xl + ((long)g * NN + j) * HID;
    float* dst = logits + ((long)row * NN + j) * HEADS;
#pragma unroll
    for (int h = 0; h < HEADS; ++h) {
      float s = 0.0f;
#pragma unroll
      for (int c = 0; c < OUTC; ++c) {
        float e = sxr[h * OUTC + c] + xlr[h * OUTC + c];
        e = (e > 0.0f) ? e : NEG_SLOPE * e;
        s += e * satt[h * OUTC + c];
      }
      dst[h] = s;
    }
  }
}

// ---------------------------------------------------------------------------
// Softmax over j (axis=1) per (g,i,h), in place. grid = ROWS, block = 256.
// ---------------------------------------------------------------------------
__global__ void softmax_j_kernel(float* __restrict__ logits) {
  const int row = blockIdx.x;
  const int tid = threadIdx.x;
  __shared__ float red[256];
  float* base = logits + (long)row * NN * HEADS;
  for (int h = 0; h < HEADS; ++h) {
    float mx = -1e30f;
    for (int j = tid; j < NN; j += 256) mx = fmaxf(mx, base[j * HEADS + h]);
    red[tid] = mx;
    __syncthreads();
    for (int s = 128; s > 0; s >>= 1) {
      if (tid < s) red[tid] = fmaxf(red[tid], red[tid + s]);
      __syncthreads();
    }
    mx = red[0];
    __syncthreads();
    float sm = 0.0f;
    for (int j = tid; j < NN; j += 256) {
      const float e = __expf(base[j * HEADS + h] - mx);
      base[j * HEADS + h] = e;
      sm += e;
    }
    red[tid] = sm;
    __syncthreads();
    for (int s = 128; s > 0; s >>= 1) {
      if (tid < s) red[tid] += red[tid + s];
      __syncthreads();
    }
    const float inv = 1.0f / red[0];
    __syncthreads();
    for (int j = tid; j < NN; j += 256) base[j * HEADS + h] *= inv;
    __syncthreads();
  }
}

// ---------------------------------------------------------------------------
// Sparsity: per-(g,i) block sums sigmoid of symmetrized generator scores.
// grid = ROWS, block = 256. Deterministic tree reduce -> partials[row].
// ---------------------------------------------------------------------------
__global__ void sparsity_kernel(const float* __restrict__ ai,
                                const float* __restrict__ aj,
                                const float* __restrict__ ab1,
                                const float* __restrict__ aw2,
                                const float* __restrict__ ab2,
                                const float* __restrict__ temp,
                                float* __restrict__ partials) {
  const int row = blockIdx.x;  // g*NN + i
  const int g = row / NN;
  const int tid = threadIdx.x;
  __shared__ float sai[GHD], saji[GHD], sab[GHD], sw[GHD];
  if (tid < GHD) {
    sai[tid] = ai[(long)row * GHD + tid];
    saji[tid] = aj[(long)row * GHD + tid];
    sab[tid] = ab1[tid];
    sw[tid] = aw2[tid];
  }
  __syncthreads();
  float t = temp[0];
  t = fminf(fmaxf(t, 0.1f), 5.0f);
  const float b2 = ab2[0];
  float acc = 0.0f;
  for (int j = tid; j < NN; j += 256) {
    const float* ajr = aj + ((long)g * NN + j) * GHD;
    const float* air = ai + ((long)g * NN + j) * GHD;
    float sij = 0.0f, sji = 0.0f;
    for (int c = 0; c < GHD; ++c) {
      float e1 = sai[c] + ajr[c] + sab[c];
      e1 = fmaxf(e1, 0.0f);
      sij += e1 * sw[c];
      float e2 = air[c] + saji[c] + sab[c];
      e2 = fmaxf(e2, 0.0f);
      sji += e2 * sw[c];
    }
    const float sc = ((sij + b2) + (sji + b2)) * 0.5f / t;
    acc += 1.0f / (1.0f + __expf(-sc));
  }
  __shared__ float red[256];
  red[tid] = acc;
  __syncthreads();
  for (int s = 128; s > 0; s >>= 1) {
    if (tid < s) red[tid] += red[tid + s];
    __syncthreads();
  }
  if (tid == 0) partials[row] = red[0];
}

__global__ void sparsity_reduce_kernel(const float* __restrict__ partials,
                                       float* __restrict__ out) {
  __shared__ float red[256];
  const int tid = threadIdx.x;
  float s = 0.0f;
  for (int k = tid; k < ROWS; k += 256) s += partials[k];
  red[tid] = s;
  __syncthreads();
  for (int st = 128; st > 0; st >>= 1) {
    if (tid < st) red[tid] += red[tid + st];
    __syncthreads();
  }
  if (tid == 0) out[BQ] = red[0] * (0.01f / ((float)BQ * NN * NN));
}

// gate scores: gs[row] = dot(tanhbuf[row], w2) + b2
__global__ void gs_kernel(const float* __restrict__ tb,
                          const float* __restrict__ w2,
                          const float* __restrict__ b2,
                          float* __restrict__ gs) {
  const int idx = blockIdx.x * blockDim.x + threadIdx.x;
  if (idx < ROWS) {
    const float* r = tb + (long)idx * HID;
    float s = 0.0f;
    for (int d = 0; d < HID; ++d) s += r[d] * w2[d];
    gs[idx] = s + b2[0];
  }
}

// gated pooling: softmax(gs) over nodes, gemb[g] = sum_i xh[g,i]*w[i]
__global__ void pool_kernel(const float* __restrict__ xh,
                            const float* __restrict__ gs,
                            float* __restrict__ gemb) {
  const int g = blockIdx.x;
  const int tid = threadIdx.x;
  __shared__ float red[256];
  float mx = -1e30f;
  for (int i = tid; i < NN; i += 256) mx = fmaxf(mx, gs[g * NN + i]);
  red[tid] = mx;
  __syncthreads();
  for (int s = 128; s > 0; s >>= 1) {
    if (tid < s) red[tid] = fmaxf(red[tid], red[tid + s]);
    __syncthreads();
  }
  mx = red[0];
  __syncthreads();
  float sm = 0.0f;
  for (int i = tid; i < NN; i += 256) sm += __expf(gs[g * NN + i] - mx);
  red[tid] = sm;
  __syncthreads();
  for (int s = 128; s > 0; s >>= 1) {
    if (tid < s) red[tid] += red[tid + s];
    __syncthreads();
  }
  const float den = red[0];
  if (tid < HID) {
    float acc = 0.0f;
    for (int i = 0; i < NN; ++i)
      acc += xh[((long)g * NN + i) * HID + tid] * __expf(gs[g * NN + i] - mx);
    gemb[g * HID + tid] = acc / den;
  }
}

// head MLP: 128->128 relu ->64 relu ->1.  grid = BQ, block = 128.
__global__ void head_kernel(const float* __restrict__ gemb,
                            const float* __restrict__ w1,
                            const float* __restrict__ b1,
                            const float* __restrict__ w2,
                            const float* __restrict__ b2,
                            const float* __restrict__ w3,
                            const float* __restrict__ b3,
                            float* __restrict__ out) {
  const int g = blockIdx.x;
  const int tid = threadIdx.x;
  __shared__ float v0[HID], v1[HID], v2[GHD];
  v0[tid] = gemb[g * HID + tid];
  __syncthreads();
  {
    const float* wr = w1 + (long)tid * HID;
    float s = 0.0f;
    for (int k = 0; k < HID; ++k) s += wr[k] * v0[k];
    v1[tid] = fmaxf(s + b1[tid], 0.0f);
  }
  __syncthreads();
  if (tid < GHD) {
    const float* wr = w2 + (long)tid * HID;
    float s = 0.0f;
    for (int k = 0; k < HID; ++k) s += wr[k] * v1[k];
    v2[tid] = fmaxf(s + b2[tid], 0.0f);
  }
  __syncthreads();
  if (tid == 0) {
    float s = 0.0f;
    for (int k = 0; k < GHD; ++k) s += w3[k] * v2[k];
    out[g] = s + b3[0];
  }
}

// ---------------------------------------------------------------------------
extern "C" void kernel_launch(void* const* d_in, const int* in_sizes, int n_in,
                              void* d_out, int out_size, void* d_ws,
                              size_t ws_size, hipStream_t stream) {
  (void)in_sizes; (void)out_size; (void)ws_size;
  const float* in[64];
  for (int i = 0; i < n_in && i < 64; ++i) in[i] = (const float*)d_in[i];
  float* out = (float*)d_out;
  float* ws = (float*)d_ws;

  // jax pytree order (dicts sorted by key):
  // 0:x  1:enc.b1 2:enc.b2 3:enc.be 4:enc.g 5:enc.w1 6:enc.w2
  // 7:gate.b1 8:gate.b2 9:gate.w1 10:gate.w2
  // 11:gg.ab1 12:gg.ab2 13:gg.aw1 14:gg.aw2 15:gg.tb1 16:gg.tb2 17:gg.temp
  // 18:gg.tw1 19:gg.tw2
  // 20-27: L0{att,bias,bl,br,ln_b,ln_g,wl,wr}
  // 28-37: L1{att,bias,bl,br,ln_b,ln_g,res_b,res_w,wl,wr}
  // 38-47: L2{...}
  // 48:proj_b 49:proj_w  50:h.b1 51:h.b2 52:h.b3 53:h.w1 54:h.w2 55:h.w3
  struct Lp { const float *att, *bias, *bl, *br, *lnb, *lng, *resb, *resw, *wl, *wr; };
  Lp L[3];
  L[0] = {in[20], in[21], in[22], in[23], in[24], in[25], nullptr, nullptr, in[26], in[27]};
  L[1] = {in[28], in[29], in[30], in[31], in[32], in[33], in[34], in[35], in[36], in[37]};
  L[2] = {in[38], in[39], in[40], in[41], in[42], in[43], in[44], in[45], in[46], in[47]};

  const long S = (long)ROWS * HID;
  float* bufA  = ws;
  float* hbuf  = ws + 1 * S;
  float* bufB  = ws + 2 * S;
  float* xh    = ws + 3 * S;
  float* xl    = ws + 4 * S;
  float* xr    = ws + 5 * S;
  float* ident = ws + 6 * S;
  float* outb  = ws + 7 * S;
  float* z1    = ws + 8 * S;
  float* zb    = z1 + (long)ROWS * GHD;
  float* ai    = zb + (long)ROWS * GHD;
  float* aj    = ai + (long)ROWS * GHD;
  float* logits = ws + 10 * S;
  float* gsb     = logits + (long)ROWS * NN * HEADS;
  float* colmean = gsb + ROWS;
  float* gemb    = colmean + BQ * HID;
  float* partials = gemb + BQ * HID;

  auto GEMM = [&](const float* A, int sAm, int sAk, const float* Bp, int sBk,
                  int sBn, const float* bias, float* C, int sCm, int M, int Nn,
                  int K, int act) {
    dim3 grid(Nn / 16, (M + 15) / 16);
    hipLaunchKernelGGL(gemm_wmma_kernel, grid, dim3(32), 0, stream,
                       A, sAm, sAk, Bp, sBk, sBn, bias, C, sCm, M, K, act);
  };

  // ---- encoder ----
  GEMM(in[0], INDIM, 1, in[5], 1, INDIM, in[1], bufA, HID, ROWS, HID, INDIM, 0);
  hipLaunchKernelGGL(lnrelu_kernel, dim3(ROWS), dim3(HID), 0, stream,
                     bufA, in[4], in[3], bufB, 1);
  GEMM(bufB, HID, 1, in[6], 1, HID, in[2], hbuf, HID, ROWS, HID, HID, 0);

  // ---- graph generator (sparsity only) ----
  GEMM(hbuf, HID, 1, in[18], 1, HID, in[15], z1, GHD, ROWS, GHD, HID, 1);
  GEMM(z1, GHD, 1, in[19], 1, GHD, in[16], zb, GHD, ROWS, GHD, GHD, 0);
  GEMM(zb, GHD, 1, in[13], 1, 2 * GHD, nullptr, ai, GHD, ROWS, GHD, GHD, 0);
  GEMM(zb, GHD, 1, in[13] + GHD, 1, 2 * GHD, nullptr, aj, GHD, ROWS, GHD, GHD, 0);
  hipLaunchKernelGGL(sparsity_kernel, dim3(ROWS), dim3(256), 0, stream,
                     ai, aj, in[11], in[14], in[12], in[17], partials);
  hipLaunchKernelGGL(sparsity_reduce_kernel, dim3(1), dim3(256), 0, stream,
                     partials, out);

  // ---- GNN ----
  GEMM(hbuf, HID, 1, in[49], 1, HID, in[48], xh, HID, ROWS, HID, HID, 0);
  for (int li = 0; li < 3; ++li) {
    if (li == 0)
      hipMemcpyAsync(ident, xh, S * sizeof(float), hipMemcpyDeviceToDevice, stream);
    else
      GEMM(xh, HID, 1, L[li].resw, 1, HID, L[li].resb, ident, HID, ROWS, HID, HID, 0);
    GEMM(xh, HID, 1, L[li].wl, 1, HID, L[li].bl, xl, HID, ROWS, HID, HID, 0);
    GEMM(xh, HID, 1, L[li].wr, 1, HID, L[li].br, xr, HID, ROWS, HID, HID, 0);
    hipLaunchKernelGGL(logits_kernel, dim3(ROWS), dim3(256), 0, stream,
                       xl, xr, L[li].att, logits);
    hipLaunchKernelGGL(softmax_j_kernel, dim3(ROWS), dim3(256), 0, stream, logits);
    hipLaunchKernelGGL(attn_agg_wmma_kernel, dim3(2, 17, BQ * HEADS), dim3(32),
                       0, stream, logits, xl, L[li].bias, outb);
    hipLaunchKernelGGL(colmean_kernel, dim3(BQ), dim3(HID), 0, stream, outb, colmean);
    hipLaunchKernelGGL(pn_res_ln_kernel, dim3(ROWS), dim3(HID), 0, stream,
                       outb, colmean, ident, L[li].lng, L[li].lnb, xh,
                       (li < 2) ? 1 : 0);
  }

  // ---- gated pooling + head ----
  GEMM(xh, HID, 1, in[9], 1, HID, in[7], bufA, HID, ROWS, HID, HID, 2);  // tanh
  hipLaunchKernelGGL(gs_kernel, dim3((ROWS + 255) / 256), dim3(256), 0, stream,
                     bufA, in[10], in[8], gsb);
  hipLaunchKernelGGL(pool_kernel, dim3(BQ), dim3(256), 0, stream, xh, gsb, gemb);
  hipLaunchKernelGGL(head_kernel, dim3(BQ), dim3(128), 0, stream,
                     gemb, in[53], in[50], in[54], in[51], in[55], in[52], out);
}